// PolyganCPlayer_54004918780527
// MI455X (gfx1250) — compile-verified
//
#include <hip/hip_runtime.h>

// ---- problem constants (match reference) ----
#define SDIM   65536          // S = 256*256
#define RANK   512
#define NMAT   8
#define CHUNKS 128            // s-chunks for phase-1 partials
#define CH     (SDIM / CHUNKS) // 512 s per chunk

typedef float v2f  __attribute__((ext_vector_type(2)));
typedef float v4f  __attribute__((ext_vector_type(4)));
typedef float v8f  __attribute__((ext_vector_type(8)));

// ---------------------------------------------------------------------------
// Phase 1: part[n1][chunk][r] = sum_{s in chunk} z[s] * W[n1+1, s, r]
// One block per (chunk, n1). 256 threads, each owns 2 consecutive r columns,
// so each wave's load is 32 lanes x 8B = 256B contiguous -> perfect coalescing.
// W is streamed once: non-temporal loads keep it from polluting L2.
// Deterministic (no atomics): each block writes its own partial row.
// ---------------------------------------------------------------------------
__global__ void __launch_bounds__(256)
k_partials(const float* __restrict__ z, const float* __restrict__ W,
           float* __restrict__ part) {
  __shared__ float zs[CH];
  const int chunk = blockIdx.x;   // 0..CHUNKS-1
  const int n1    = blockIdx.y;   // 0..6  -> matrix W[n1+1]
  const int t     = threadIdx.x;  // 0..255
  const int s0    = chunk * CH;

  zs[t]       = z[s0 + t];
  zs[t + 256] = z[s0 + t + 256];
  __syncthreads();

  const int r = t * 2;
  const float* wp = W + ((size_t)(n1 + 1) * SDIM + (size_t)s0) * RANK + r;
  float acc0 = 0.0f, acc1 = 0.0f;

#pragma unroll 4
  for (int s = 0; s < CH; ++s) {
    v2f w = __builtin_nontemporal_load((const v2f*)wp);  // TH=NT streaming
    const float zz = zs[s];
    acc0 = fmaf(zz, w.x, acc0);
    acc1 = fmaf(zz, w.y, acc1);
    wp += RANK;
  }

  float* pp = part + ((size_t)(n1 * CHUNKS + chunk)) * RANK + r;
  pp[0] = acc0;
  pp[1] = acc1;
}

// ---------------------------------------------------------------------------
// Phase 2a: G[n1][r] = sum over chunks of part[n1][chunk][r]   (7 blocks x 512)
// ---------------------------------------------------------------------------
__global__ void __launch_bounds__(512)
k_reduceG(const float* __restrict__ part, float* __restrict__ G) {
  const int n1 = blockIdx.x;   // 0..6
  const int r  = threadIdx.x;  // 0..511
  const float* pp = part + (size_t)n1 * CHUNKS * RANK + r;
  float acc = 0.0f;
#pragma unroll 8
  for (int c = 0; c < CHUNKS; ++c) {
    acc += pp[0];
    pp += RANK;
  }
  G[n1 * RANK + r] = acc;
}

// ---------------------------------------------------------------------------
// Phase 2b: Ssum[r] = 2 + sum_{i=0..5} prod_{j<=i} G[j][r]
// (F = [1,1,cp0..cp5]; column sum of F)
// ---------------------------------------------------------------------------
__global__ void __launch_bounds__(512)
k_ssum(const float* __restrict__ G, float* __restrict__ Ssum) {
  const int r = threadIdx.x;  // 0..511
  float p = 1.0f, s = 2.0f;
#pragma unroll
  for (int i = 0; i < 6; ++i) {
    p *= G[i * RANK + r];
    s += p;
  }
  Ssum[r] = s;
}

// ---------------------------------------------------------------------------
// Phase 3: out[s] = b[s] + W[0,s,:] . Ssum   via V_WMMA_F32_16X16X4_F32.
//
// Per WMMA (D = A x B + C, 16x16x4 f32):
//   A[M][K] = W0[s0+M][k0+K]  -- documented A layout: lanes 0-15 hold (K=0,K=1)
//             in (v0,v1), lanes 16-31 hold (K=2,K=3) => one contiguous b64
//             load per lane; the wave consumes exactly the 16x4 tile (256B).
//   B[K][N] = Ssum[k0+K] replicated across all N (broadcast b64 from LDS).
// => D[M][N] = partial dot for row s0+M, identical across N.
// C/D layout: VGPR j = row M=j (lanes 0-15) / M=j+8 (lanes 16-31), so lane 0
// holds rows s0..s0+7 and lane 16 holds rows s0+8..s0+15; they add bias and
// store float4s. 256 threads = 8 waves, each wave owns a 16-row tile.
// ---------------------------------------------------------------------------
__global__ void __launch_bounds__(256)
k_out_wmma(const float* __restrict__ W, const float* __restrict__ Ssum,
           const float* __restrict__ b, float* __restrict__ out) {
  __shared__ float ss[RANK];
  const int t = threadIdx.x;  // 0..255
  ss[t]       = Ssum[t];
  ss[t + 256] = Ssum[t + 256];
  __syncthreads();

  const int wave = t >> 5;
  const int lane = t & 31;
  const int M    = lane & 15;
  const int kg   = lane >> 4;                    // 0: K=0,1   1: K=2,3
  const int s0   = blockIdx.x * 128 + wave * 16; // 16-row tile per wave

  const float* ap = W + (size_t)(s0 + M) * RANK + (kg * 2);  // W[0] rows
  const float* bp = ss + (kg * 2);

  v8f c = {};
#pragma unroll 4
  for (int k0 = 0; k0 < RANK; k0 += 4) {
    v2f a  = *(const v2f*)ap;   // A fragment: 2 consecutive K of row s0+M
    v2f bb = *(const v2f*)bp;   // B fragment: Ssum[k0+kg*2 .. +1], bcast over N
    c = __builtin_amdgcn_wmma_f32_16x16x4_f32(
        /*neg_a=*/false, a, /*neg_b=*/false, bb,
        /*c_mod=*/(short)0, c, /*reuse_a=*/false, /*reuse_b=*/false);
    ap += 4;
    bp += 4;
  }

  // N-columns are identical; lane 0 -> rows s0..s0+7, lane 16 -> s0+8..s0+15
  if ((lane & 15) == 0) {
    const int row = s0 + (kg ? 8 : 0);
    v4f b0 = *(const v4f*)(b + row);
    v4f b1 = *(const v4f*)(b + row + 4);
    v4f o0 = { c[0] + b0.x, c[1] + b0.y, c[2] + b0.z, c[3] + b0.w };
    v4f o1 = { c[4] + b1.x, c[5] + b1.y, c[6] + b1.z, c[7] + b1.w };
    *(v4f*)(out + row)     = o0;
    *(v4f*)(out + row + 4) = o1;
  }
}

// ---------------------------------------------------------------------------
// Launch: 4 kernels on one stream (sequential dependency).
// Workspace layout (floats): part[7*128*512] | G[7*512] | Ssum[512] (~1.85 MB)
// ---------------------------------------------------------------------------
extern "C" void kernel_launch(void* const* d_in, const int* in_sizes, int n_in,
                              void* d_out, int out_size, void* d_ws, size_t ws_size,
                              hipStream_t stream) {
  const float* z = (const float*)d_in[0];          // (S,)
  const float* W = (const float*)d_in[1];          // (N, S, RANK)
  const float* b = (const float*)d_in[2];          // (S,)
  float* out = (float*)d_out;                      // (S,)

  float* part = (float*)d_ws;                      // 7*CHUNKS*RANK
  float* G    = part + (size_t)7 * CHUNKS * RANK;  // 7*RANK
  float* Ss   = G + 7 * RANK;                      // RANK

  k_partials<<<dim3(CHUNKS, 7), 256, 0, stream>>>(z, W, part);
  k_reduceG <<<7, 512, 0, stream>>>(part, G);
  k_ssum    <<<1, 512, 0, stream>>>(G, Ss);
  k_out_wmma<<<SDIM / 128, 256, 0, stream>>>(W, Ss, b, out);
}